// WaveLetPooling_81561428951816
// MI455X (gfx1250) — compile-verified
//
#include <hip/hip_runtime.h>
#include <hip/hip_bf16.h>

// WaveLetPooling: (8,512,512,64) f32 -> 4 x (8,256,256,64) f32 (ll,lh,hl,hh).
// Memory-bound (AI ~0.6 flop/byte). Strategy: CDNA5 async global->LDS staging
// (ASYNCcnt path), b128 everywhere, non-temporal stores.

typedef float v4f __attribute__((ext_vector_type(4)));

#define BATCH 8
#define H     512
#define W     512
#define C     64
#define OH    (H / 2)
#define OW    (W / 2)
#define PX_PER_BLOCK 16          // output pixels per block along ox
#define THREADS 256

__global__ __launch_bounds__(THREADS)
void wavelet_pool_kernel(const float* __restrict__ in,
                         float* __restrict__ oll,
                         float* __restrict__ olh,
                         float* __restrict__ ohl,
                         float* __restrict__ ohh) {
    // 16 KiB staging: [0,2048) floats = row y0 chunk (32 px * 64 ch),
    //                 [2048,4096)    = row y1 chunk.
    __shared__ float sh[4096];

    const int      tid    = threadIdx.x;
    const unsigned bid    = blockIdx.x;          // 32768 blocks
    const unsigned oxBlk  = bid & 15u;           // 16 blocks cover one ox row
    const unsigned rowIdx = bid >> 4;            // batch*256 + oy
    const unsigned oy     = rowIdx & 255u;
    const unsigned batch  = rowIdx >> 8;
    const unsigned ox0    = oxBlk * PX_PER_BLOCK;

    const size_t rowStride = (size_t)W * C;                              // 32768 floats
    const size_t base0 = (((size_t)batch * H + 2u * oy) * W + 2u * ox0) * C;
    const float* g0 = in + base0;                // 8 KiB contiguous (row y0)
    const float* g1 = g0 + rowStride;            // 8 KiB contiguous (row y1)

    // ---- async stage 16 KiB global -> LDS (CDNA5 async-to-LDS path) ----
    {
        const unsigned ldsBase = (unsigned)(size_t)(void*)&sh[0]; // low 32 bits = LDS offset
        const unsigned la      = ldsBase + (unsigned)tid * 16u;
        const float*   ga = g0 + tid * 4;          // bytes [tid*16, tid*16+16) of row0
        const float*   gb = g0 + 1024 + tid * 4;   // second 4 KiB of row0
        const float*   gc = g1 + tid * 4;
        const float*   gd = g1 + 1024 + tid * 4;
        asm volatile("global_load_async_to_lds_b128 %0, %1, off"
                     :: "v"(la),           "v"(ga) : "memory");
        asm volatile("global_load_async_to_lds_b128 %0, %1, off"
                     :: "v"(la + 4096u),   "v"(gb) : "memory");
        asm volatile("global_load_async_to_lds_b128 %0, %1, off"
                     :: "v"(la + 8192u),   "v"(gc) : "memory");
        asm volatile("global_load_async_to_lds_b128 %0, %1, off"
                     :: "v"(la + 12288u),  "v"(gd) : "memory");
    }
#if __has_builtin(__builtin_amdgcn_s_wait_asynccnt)
    __builtin_amdgcn_s_wait_asynccnt(0);
#else
    asm volatile("s_wait_asynccnt 0" ::: "memory");
#endif
    __syncthreads();

    // ---- butterfly: thread t -> output pixel (t>>4), channel quad (t&15) ----
    const int px = tid >> 4;                    // 0..15
    const int c4 = tid & 15;                    // 0..15 (4 channels each)
    const int fo = c4 * 4;

    const v4f a = *(const v4f*)&sh[(2 * px    ) * C + fo];          // (y0, 2x)
    const v4f b = *(const v4f*)&sh[(2 * px + 1) * C + fo];          // (y0, 2x+1)
    const v4f c = *(const v4f*)&sh[2048 + (2 * px    ) * C + fo];   // (y1, 2x)
    const v4f d = *(const v4f*)&sh[2048 + (2 * px + 1) * C + fo];   // (y1, 2x+1)

    const v4f e = a + b;
    const v4f f = a - b;
    const v4f g = c + d;
    const v4f h = c - d;

    const v4f ll =  0.5f * (e + g);   //  0.5*( a+b+c+d)
    const v4f lh = -0.5f * (f + h);   //  0.5*(-a+b-c+d)
    const v4f hl =  0.5f * (g - e);   //  0.5*(-a-b+c+d)
    const v4f hh =  0.5f * (f - h);   //  0.5*( a-b-c+d)

    const size_t opix = ((size_t)batch * OH + oy) * OW + (ox0 + px);
    const size_t off  = opix * C + fo;

    __builtin_nontemporal_store(ll, (v4f*)(oll + off));
    __builtin_nontemporal_store(lh, (v4f*)(olh + off));
    __builtin_nontemporal_store(hl, (v4f*)(ohl + off));
    __builtin_nontemporal_store(hh, (v4f*)(ohh + off));
}

extern "C" void kernel_launch(void* const* d_in, const int* in_sizes, int n_in,
                              void* d_out, int out_size, void* d_ws, size_t ws_size,
                              hipStream_t stream) {
    const float* in  = (const float*)d_in[0];
    float*       out = (float*)d_out;
    const size_t quarter = (size_t)BATCH * OH * OW * C;   // 33,554,432 floats per output

    float* oll = out;
    float* olh = out + quarter;
    float* ohl = out + 2 * quarter;
    float* ohh = out + 3 * quarter;

    const unsigned nPix   = (unsigned)(BATCH * OH * OW);  // 524288 output pixels
    const unsigned blocks = nPix / PX_PER_BLOCK;          // 32768

    wavelet_pool_kernel<<<blocks, THREADS, 0, stream>>>(in, oll, olh, ohl, ohh);
}